// PhysicsInformedLoss_3642132267418
// MI455X (gfx1250) — compile-verified
//
#include <hip/hip_runtime.h>
#include <hip/hip_bf16.h>

// ---------------------------------------------------------------------------
// PhysicsInformedLoss for MI455X (gfx1250, wave32).
// Memory-bound scatter/reduce. Roofline: ~180 MB HBM traffic -> ~8 us floor
// at 23.3 TB/s; compute is noise. Hot L2 working set = V/theta gathers
// (2.5 MB) + calc_pf/calc_reac atomic targets (2.56 MB) << 192 MB L2, so
// read-once streams are loaded non-temporally to protect it.
// WMMA f32 16x16x4 (ones-matrix trick) is the full-precision wave-sum
// primitive: one matrix op replaces a 5-step shuffle tree.
// ---------------------------------------------------------------------------

typedef __attribute__((ext_vector_type(2))) float v2f;
typedef __attribute__((ext_vector_type(8))) float v8f;

#define NTLOAD(p) __builtin_nontemporal_load(p)

// Full-precision wave32 sum using V_WMMA_F32_16X16X4_F32 with B = ones.
// A is 16x4 f32 (2 VGPRs/lane): VGPR0 holds the lane partial, VGPR1 zero, so
// each lane's value appears exactly once in A. With B[k][n]=1, D rows are
// rowsum[m] = x[m] + x[m+16]; summing the 8 D registers yields S_low in lanes
// 0-15 and S_high in lanes 16-31; one shfl_xor(16) completes the wave sum in
// every lane. EXEC must be all-1s in the calling wave.
__device__ __forceinline__ float wave_reduce_wmma(float x) {
    v2f a; a.x = x;    a.y = 0.0f;
    v2f b; b.x = 1.0f; b.y = 1.0f;
    v8f c = {};
    c = __builtin_amdgcn_wmma_f32_16x16x4_f32(
            /*neg_a=*/false, a, /*neg_b=*/false, b,
            /*c_mod=*/(short)0, c, /*reuse_a=*/false, /*reuse_b=*/false);
    float s = ((c[0] + c[1]) + (c[2] + c[3])) + ((c[4] + c[5]) + (c[6] + c[7]));
    s += __shfl_xor(s, 16, 32);
    return s;
}

// Block (256 threads = 8 waves) reduction:
// WMMA wave sums -> LDS -> wave 0 WMMA-reduces the 8 partials -> 1 atomic.
__device__ __forceinline__ void block_reduce_atomic_256(float x, float* dest) {
    __shared__ float wsum[8];
    const int lane = threadIdx.x & 31;
    const int wid  = threadIdx.x >> 5;
    const float s = wave_reduce_wmma(x);
    if (lane == 0) wsum[wid] = s;
    __syncthreads();
    if (wid == 0) {                       // whole wave 0 active: EXEC all-1s
        const float t = (lane < 8) ? wsum[lane] : 0.0f;
        const float bsum = wave_reduce_wmma(t);
        if (lane == 0) atomicAdd(dest, bsum);
    }
}

// ---------------------------------------------------------------------------
// Edge kernel: one thread per edge, loop over batches. Computes active (pf)
// and reactive flows, scatter-adds into calc_pf / calc_reac (L2-resident),
// and accumulates the thermal-capacity violation sum.
// Streamed read-once data uses NT loads; gathered V/theta use cached loads.
// ---------------------------------------------------------------------------
__global__ void __launch_bounds__(256)
edge_kernel(const int* __restrict__ eidx,
            const float* __restrict__ volts,  const float* __restrict__ angles,
            const float* __restrict__ cond,   const float* __restrict__ susc,
            const float* __restrict__ flows,  const float* __restrict__ tlim,
            float* __restrict__ calc_pf, float* __restrict__ calc_reac,
            float* __restrict__ accum_cap, int B, int N, int E) {
    const int e = blockIdx.x * blockDim.x + threadIdx.x;
    float cap_local = 0.0f;
    if (e < E) {
        const int src = NTLOAD(&eidx[e]);
        const int dst = NTLOAD(&eidx[E + e]);
        #pragma unroll 4
        for (int b = 0; b < B; ++b) {
            const int nb = b * N;
            const int eb = b * E + e;
            const float Vi = volts[nb + src];          // hot, cached
            const float Vj = volts[nb + dst];
            const float thij = angles[nb + src] - angles[nb + dst];
            float sn, cs;
            __sincosf(thij, &sn, &cs);
            const float G  = NTLOAD(&cond[eb]);        // streamed, read-once
            const float Bs = NTLOAD(&susc[eb]);
            const float VV = Vi * Vj;
            const float Fpf = VV * (G * cs + Bs * sn); // active flow
            const float Fre = VV * (G * sn - Bs * cs); // reactive flow
            atomicAdd(&calc_pf[nb + src],   Fpf);
            atomicAdd(&calc_pf[nb + dst],  -Fpf);
            atomicAdd(&calc_reac[nb + src], Fre);
            atomicAdd(&calc_reac[nb + dst], -Fre);
            const float viol = fmaxf(fabsf(NTLOAD(&flows[eb])) - NTLOAD(&tlim[eb]), 0.0f);
            cap_local = fmaf(viol, viol, cap_local);
        }
    }
    block_reduce_atomic_256(cap_local, accum_cap);   // all threads reconverged
}

// ---------------------------------------------------------------------------
// Node kernel: BCE + pf/reac mismatch + stability + voltage-tracking terms.
// All are means over B*N, so they fold into one weighted accumulator.
// Everything here is read exactly once -> NT loads throughout.
// ---------------------------------------------------------------------------
__global__ void __launch_bounds__(256)
node_kernel(const float* __restrict__ fprob, const float* __restrict__ flabel,
            const float* __restrict__ volts, const float* __restrict__ vtgt,
            const float* __restrict__ pinj,  const float* __restrict__ qinj,
            const float* __restrict__ calc_pf, const float* __restrict__ calc_reac,
            float* __restrict__ accum_node, int total) {
    const int i = blockIdx.x * blockDim.x + threadIdx.x;
    float contrib = 0.0f;
    if (i < total) {
        const float eps = 1e-7f;
        const float p = fminf(fmaxf(NTLOAD(&fprob[i]), eps), 1.0f - eps);
        const float y = NTLOAD(&flabel[i]);
        const float bce = -(y * logf(p) + (1.0f - y) * log1pf(-p));

        const float V    = NTLOAD(&volts[i]);
        const float low  = fmaxf(0.95f - V, 0.0f);
        const float high = fmaxf(V - 1.05f, 0.0f);
        const float stab = low * low + high * high;

        const float dv  = V - NTLOAD(&vtgt[i]);
        const float dpf = NTLOAD(&calc_pf[i])   - NTLOAD(&pinj[i]);
        const float dre = NTLOAD(&calc_reac[i]) - NTLOAD(&qinj[i]);

        // bce + LAM_PF*pf + LAM_REAC*reac + LAM_STAB*stab + LAM_VOLT*volt
        contrib = bce
                + 0.10f * (dpf * dpf)
                + 0.10f * (dre * dre)
                + 0.05f * stab
                + 1.00f * (dv * dv);
    }
    block_reduce_atomic_256(contrib, accum_node);
}

// ---------------------------------------------------------------------------
// Final kernel: one wave. Lanes 0..B-1 compute the frequency-deviation terms;
// WMMA wave reduce; lane 0 composes the weighted total loss.
// ---------------------------------------------------------------------------
__global__ void __launch_bounds__(32)
final_kernel(const float* __restrict__ freq, const float* __restrict__ pimb,
             const float* __restrict__ accum, float* __restrict__ out,
             int B, int N, int E) {
    const int l = threadIdx.x;
    float t = 0.0f;
    if (l < B) {
        // expected = 60 + imbalance/(2*5) * 60 = 60 + imbalance*6
        const float ef = fmaf(pimb[l], 6.0f, 60.0f);
        const float d  = freq[l] - ef;
        t = d * d;
    }
    const float fsum = wave_reduce_wmma(t);
    if (l == 0) {
        const float inv_bn = 1.0f / (float)(B * N);
        const float inv_be = 1.0f / ((float)B * (float)E);
        const float Lnode = accum[0] * inv_bn;                 // weighted node losses
        const float Lcap  = accum[1] * inv_be;                 // capacity loss
        const float Lfreq = fsum / (float)B;                   // frequency loss
        out[0] = Lnode + 0.05f * Lcap + 0.08f * Lfreq;
    }
}

// ---------------------------------------------------------------------------
// Launch
// ---------------------------------------------------------------------------
extern "C" void kernel_launch(void* const* d_in, const int* in_sizes, int n_in,
                              void* d_out, int out_size, void* d_ws, size_t ws_size,
                              hipStream_t stream) {
    const float* fprob  = (const float*)d_in[0];   // (B,N)
    const float* flabel = (const float*)d_in[1];   // (B,N)
    const float* volts  = (const float*)d_in[2];   // (B,N,1)
    const float* angles = (const float*)d_in[3];   // (B,N,1)
    const float* flows  = (const float*)d_in[4];   // (B,E,1)
    const float* freq   = (const float*)d_in[5];   // (B,1)
    const float* vtgt   = (const float*)d_in[6];   // (B,N,1)
    const float* pinj   = (const float*)d_in[7];   // (B,N,1)
    const float* qinj   = (const float*)d_in[8];   // (B,N,1)
    const float* cond   = (const float*)d_in[9];   // (B,E)
    const float* susc   = (const float*)d_in[10];  // (B,E)
    const float* tlim   = (const float*)d_in[11];  // (B,E)
    const float* pimb   = (const float*)d_in[12];  // (B,1)
    const int*   eidx   = (const int*)d_in[13];    // (2,E)

    const int B = in_sizes[5];            // 16
    const int N = in_sizes[0] / B;        // 20000
    const int E = in_sizes[13] / 2;       // 640000

    // Workspace: accum[0]=node sum, accum[1]=cap sum, then calc_pf, calc_reac.
    float* accum     = (float*)d_ws;
    float* calc_pf   = accum + 64;        // 256B-aligned region
    float* calc_reac = calc_pf + (size_t)B * N;
    const size_t zero_bytes = (size_t)(64 + 2 * (size_t)B * N) * sizeof(float);
    hipMemsetAsync(d_ws, 0, zero_bytes, stream);   // graph-capturable memset node

    const int TB = 256;
    edge_kernel<<<(E + TB - 1) / TB, TB, 0, stream>>>(
        eidx, volts, angles, cond, susc, flows, tlim,
        calc_pf, calc_reac, accum + 1, B, N, E);

    node_kernel<<<(B * N + TB - 1) / TB, TB, 0, stream>>>(
        fprob, flabel, volts, vtgt, pinj, qinj,
        calc_pf, calc_reac, accum, B * N);

    final_kernel<<<1, 32, 0, stream>>>(freq, pimb, accum, (float*)d_out, B, N, E);
}